// PatchPooling_46952582480042
// MI455X (gfx1250) — compile-verified
//
#include <hip/hip_runtime.h>
#include <hip/hip_bf16.h>
#include <stdint.h>

typedef float v2f  __attribute__((ext_vector_type(2)));
typedef float v4fv __attribute__((ext_vector_type(4)));
typedef float v8f  __attribute__((ext_vector_type(8)));
typedef int   v4i  __attribute__((ext_vector_type(4)));

#define B_  16
#define S_  8192
#define D_  256
#define P_  256
#define PT  16                    // patches per workgroup (WMMA M)
#define NW  16                    // waves per workgroup (one per 16-wide D tile)
#define THREADS (32 * NW)
#define CHUNK 24                  // rows per staging buffer (multiple of 4)
#define ROWF  264                 // padded row pitch in floats (1056 B, 16B aligned,
                                  // pitch mod 64 banks = 8 -> conflict-free halves)
#define LDSF  (CHUNK * ROWF)
#define STG_IT ((CHUNK * (D_ / 4)) / THREADS)   // 24*64/512 = 3 b128 loads/thread

#if defined(__gfx1250__) && __has_builtin(__builtin_amdgcn_global_load_async_to_lds_b128)
#define USE_ASYNC 1
#define ASYNC_WAIT(n) asm volatile("s_wait_asynccnt %0" ::"n"(n) : "memory")
typedef __attribute__((address_space(1))) v4i* gas_v4i_ptr;
typedef __attribute__((address_space(3))) v4i* lds_v4i_ptr;
#else
#define USE_ASYNC 0
#define ASYNC_WAIT(n)
#endif

// ---------------------------------------------------------------------------
// Kernel 1: per-batch inclusive scan of patch_lengths -> patch ends
// ---------------------------------------------------------------------------
__global__ __launch_bounds__(P_) void PatchPool_scan_kernel(
    const int* __restrict__ lengths, int* __restrict__ ends) {
  __shared__ int sh[P_];
  const int b = blockIdx.x;
  const int t = threadIdx.x;
  sh[t] = lengths[b * P_ + t];
  __syncthreads();
  #pragma unroll
  for (int off = 1; off < P_; off <<= 1) {
    int v = (t >= off) ? sh[t - off] : 0;
    __syncthreads();
    sh[t] += v;
    __syncthreads();
  }
  ends[b * P_ + t] = sh[t];
}

// ---------------------------------------------------------------------------
// Kernel 2: async-LDS-staged masked segment-sum via V_WMMA_F32_16X16X4_F32
//   grid = (P_/PT, B_), block = 512 (16 waves). Double-buffered CHUNK-row
//   staging (global_load_async_to_lds_b128 / ASYNCcnt) feeding 6 unrolled
//   WMMAs per chunk per wave.
// ---------------------------------------------------------------------------
__global__ __launch_bounds__(THREADS) void PatchPool_wmma_kernel(
    const float* __restrict__ batch, const int* __restrict__ lengths,
    const int* __restrict__ ends, float* __restrict__ out) {
  const int b  = blockIdx.y;
  const int p0 = blockIdx.x * PT;

  __shared__ int   sBeg[PT];
  __shared__ int   sEnd[PT];
  __shared__ float smem[2 * LDSF];

  const int t = threadIdx.x;
  if (t < PT) {
    const int e = ends[b * P_ + p0 + t];
    const int l = lengths[b * P_ + p0 + t];
    sBeg[t] = e - l;
    sEnd[t] = e;
  }
  __syncthreads();

  const int segBeg = sBeg[0];
  const int segEnd = sEnd[PT - 1];   // <= 7936, so +CHUNK-1 overread stays in S

  const int wave = t >> 5;
  const int lane = t & 31;
  const int half = lane >> 4;        // K-half of A/B fragments
  const int lm   = lane & 15;        // M index (A) / N index (B)
  const int n0   = wave * 16;

  const int myBeg = sBeg[lm];
  const int myEnd = sEnd[lm];

  const float* __restrict__ bb = batch + (size_t)b * S_ * D_;

  // Stage CHUNK rows (full width) into LDS buffer (ch & 1).
  auto stage = [&](int ch) {
    const int s0 = segBeg + ch * CHUNK;
    float* dst = &smem[(ch & 1) ? LDSF : 0];
    #pragma unroll
    for (int i = 0; i < STG_IT; ++i) {
      const int e   = t + i * THREADS;
      const int row = e >> 6;          // 64 x 16B per row
      const int col = (e & 63) << 2;   // float index
      const float* g = bb + (size_t)(s0 + row) * D_ + col;
      float*       l = dst + row * ROWF + col;
#if USE_ASYNC
      __builtin_amdgcn_global_load_async_to_lds_b128(
          (gas_v4i_ptr)(uintptr_t)g,
          (lds_v4i_ptr)(uint32_t)(uintptr_t)l,
          /*offset=*/0, /*cpol=*/0);
#else
      *(v4fv*)l = *(const v4fv*)g;
#endif
    }
  };

  v8f acc = {0.f, 0.f, 0.f, 0.f, 0.f, 0.f, 0.f, 0.f};

  const int nc = (segEnd > segBeg) ? ((segEnd - segBeg + CHUNK - 1) / CHUNK) : 0;

  if (nc > 0) stage(0);

  for (int ch = 0; ch < nc; ++ch) {
    if (ch + 1 < nc) {
      stage(ch + 1);
      ASYNC_WAIT(STG_IT);     // chunk ch landed; chunk ch+1 still in flight
    } else {
      ASYNC_WAIT(0);
    }
    __syncthreads();

    const float* sb = &smem[(ch & 1) ? LDSF : 0];
    const int    s0 = segBeg + ch * CHUNK;

    #pragma unroll
    for (int j = 0; j < CHUNK / 4; ++j) {
      const int kl = 4 * j + 2 * half;   // row-in-chunk for this lane-half
      const int k0 = s0 + kl;

      v2f a;                              // 16x4 {0,1} membership mask
      a.x = (k0     >= myBeg && k0     < myEnd) ? 1.0f : 0.0f;
      a.y = (k0 + 1 >= myBeg && k0 + 1 < myEnd) ? 1.0f : 0.0f;

      v2f bf;                             // 4x16 slab from LDS
      bf.x = sb[kl * ROWF + n0 + lm];
      bf.y = sb[(kl + 1) * ROWF + n0 + lm];

      acc = __builtin_amdgcn_wmma_f32_16x16x4_f32(
          /*neg_a=*/false, a, /*neg_b=*/false, bf,
          /*c_mod=*/(short)0, acc, /*reuse_a=*/false, /*reuse_b=*/false);
    }
    __syncthreads();   // protect buffer (ch & 1) before stage(ch + 2)
  }

  // C/D layout: VGPR r, lane-half h -> M = r + 8*h, N = lane & 15.
  #pragma unroll
  for (int r = 0; r < 8; ++r) {
    const int m   = r + 8 * half;
    const int len = sEnd[m] - sBeg[m];
    const float flen = (float)((len > 0) ? len : 1);
#if defined(__gfx1250__) && __has_builtin(__builtin_amdgcn_rcpf)
    const float inv = __builtin_amdgcn_rcpf(flen);
#else
    const float inv = 1.0f / flen;
#endif
    out[((size_t)b * P_ + p0 + m) * D_ + n0 + lm] = acc[r] * inv;
  }
}

// ---------------------------------------------------------------------------
extern "C" void kernel_launch(void* const* d_in, const int* in_sizes, int n_in,
                              void* d_out, int out_size, void* d_ws, size_t ws_size,
                              hipStream_t stream) {
  (void)in_sizes; (void)n_in; (void)out_size; (void)ws_size;
  const float* batch   = (const float*)d_in[0];   // [B, S, D] f32
  const int*   lengths = (const int*)d_in[1];     // [B, P] i32
  float*       out     = (float*)d_out;           // [B, P, D] f32
  int*         ends    = (int*)d_ws;              // [B, P] i32 scratch

  PatchPool_scan_kernel<<<dim3(B_), dim3(P_), 0, stream>>>(lengths, ends);
  PatchPool_wmma_kernel<<<dim3(P_ / PT, B_), dim3(THREADS), 0, stream>>>(
      batch, lengths, ends, out);
}